// GNN_Multi_Layer_54494545052310
// MI455X (gfx1250) — compile-verified
//
#include <hip/hip_runtime.h>

typedef float v2f __attribute__((ext_vector_type(2)));
typedef float v8f __attribute__((ext_vector_type(8)));

#define D 32  // D_IN == D_HID == D_OUT == 32

// ---------------------------------------------------------------------------
// Degree: deg[i] = 1 (self loop) + sum_{e: dst[e]==i} w[e]
// ---------------------------------------------------------------------------
__global__ void k_init_deg(float* __restrict__ deg, int n) {
  int i = blockIdx.x * blockDim.x + threadIdx.x;
  if (i < n) deg[i] = 1.0f;  // self-loop weight
}

__global__ void k_deg_edges(const int* __restrict__ dst, const float* __restrict__ w,
                            float* __restrict__ deg, int E) {
  int e = blockIdx.x * blockDim.x + threadIdx.x;
  if (e < E) atomicAdd(&deg[dst[e]], w[e]);
}

__global__ void k_dis(const float* __restrict__ deg, float* __restrict__ dis, int n) {
  int i = blockIdx.x * blockDim.x + threadIdx.x;
  if (i < n) {
    float d = deg[i];
    dis[i] = (d > 0.0f) ? __frsqrt_rn(d) : 0.0f;
  }
}

// ---------------------------------------------------------------------------
// Dense transform Y[N,32] = (RELU? max(X,0) : X)[N,32] @ W[32,32] via
// V_WMMA_F32_16X16X4_F32. One wave -> 16 rows x 32 cols (two 16x16 C tiles).
// Layouts per CDNA5 ISA 7.12.2:
//   A 16x4 f32 : lanes 0-15 hold M=lane, v[0]=K0 v[1]=K1; lanes 16-31 K2/K3
//   B 4x16 f32 : lanes 0-15 hold N=lane, v[0]=K0 v[1]=K1; lanes 16-31 K2/K3
//   C 16x16 f32: VGPR j, lanes 0-15 -> M=j, lanes 16-31 -> M=j+8, N=lane%16
// N_NODES = 100000 is a multiple of 16, so tiles are always full and the
// wave-uniform early-out keeps EXEC all-ones for every executed WMMA.
// ---------------------------------------------------------------------------
template <bool RELU>
__global__ void k_gemm32(const float* __restrict__ X, const float* __restrict__ W,
                         float* __restrict__ Y, int n) {
  const int lane = threadIdx.x & 31;
  const int wave = threadIdx.x >> 5;
  const int row0 = (blockIdx.x * (blockDim.x >> 5) + wave) << 4;  // 16 rows/wave
  if (row0 >= n) return;  // wave-uniform: EXEC remains all-1s below

  const int l    = lane & 15;
  const int half = lane >> 4;  // selects K pair within a K=4 step

  v8f c0 = {};  // cols 0..15
  v8f c1 = {};  // cols 16..31

  const float* xrow = X + (size_t)(row0 + l) * D;

#pragma unroll
  for (int kb = 0; kb < D; kb += 4) {
    const int k0 = kb + 2 * half;
    v2f a;
    a.x = xrow[k0];
    a.y = xrow[k0 + 1];
    if (RELU) {
      a.x = fmaxf(a.x, 0.0f);
      a.y = fmaxf(a.y, 0.0f);
    }
    v2f b0, b1;
    b0.x = W[(size_t)k0 * D + l];
    b0.y = W[(size_t)(k0 + 1) * D + l];
    b1.x = W[(size_t)k0 * D + 16 + l];
    b1.y = W[(size_t)(k0 + 1) * D + 16 + l];

    c0 = __builtin_amdgcn_wmma_f32_16x16x4_f32(false, a, false, b0, (short)0, c0,
                                               false, false);
    c1 = __builtin_amdgcn_wmma_f32_16x16x4_f32(false, a, false, b1, (short)0, c1,
                                               false, false);
  }

  const int rowm = row0 + 8 * half;
#pragma unroll
  for (int j = 0; j < 8; ++j) {
    Y[(size_t)(rowm + j) * D + l]      = c0[j];
    Y[(size_t)(rowm + j) * D + 16 + l] = c1[j];
  }
}

// ---------------------------------------------------------------------------
// out[i,d] = b[d] + xm[i,d] * dis[i]^2   (self-loop message + bias; also the
// deterministic initialization of the accumulation buffer)
// ---------------------------------------------------------------------------
__global__ void k_self_bias(const float* __restrict__ xm, const float* __restrict__ dis,
                            const float* __restrict__ b, float* __restrict__ out,
                            long long total) {
  long long idx = (long long)blockIdx.x * blockDim.x + threadIdx.x;
  if (idx >= total) return;
  int i = (int)(idx >> 5);
  int d = (int)(idx & (D - 1));
  float s = dis[i];
  out[idx] = b[d] + xm[idx] * (s * s);
}

// ---------------------------------------------------------------------------
// Edge messages: out[dst,:] += xm[src,:] * dis[src]*w*dis[dst]
// One lane per (edge, feature): 32 lanes of a wave share one edge, so the
// scalar loads broadcast and the row accesses/atomics are coalesced.
// ---------------------------------------------------------------------------
__global__ void k_edge_msg(const int* __restrict__ src, const int* __restrict__ dst,
                           const float* __restrict__ w, const float* __restrict__ dis,
                           const float* __restrict__ xm, float* __restrict__ out,
                           int E) {
  long long idx = (long long)blockIdx.x * blockDim.x + threadIdx.x;
  int e = (int)(idx >> 5);
  int d = (int)(idx & (D - 1));
  if (e >= E) return;

  if (d == 0 && e + 4096 < E) {  // stream-ahead on the HBM-resident edge lists
    __builtin_prefetch(&src[e + 4096], 0, 0);
    __builtin_prefetch(&dst[e + 4096], 0, 0);
    __builtin_prefetch(&w[e + 4096], 0, 0);
  }

  int s = src[e];
  int t = dst[e];
  float c = dis[s] * w[e] * dis[t];
  atomicAdd(&out[(size_t)t * D + d], xm[(size_t)s * D + d] * c);
}

// ---------------------------------------------------------------------------
// Host-side orchestration
// ---------------------------------------------------------------------------
extern "C" void kernel_launch(void* const* d_in, const int* in_sizes, int n_in,
                              void* d_out, int out_size, void* d_ws, size_t ws_size,
                              hipStream_t stream) {
  const float* x  = (const float*)d_in[0];
  const int*   ei = (const int*)d_in[1];
  const float* ew = (const float*)d_in[2];
  const float* W1 = (const float*)d_in[3];
  const float* b1 = (const float*)d_in[4];
  const float* W2 = (const float*)d_in[5];
  const float* b2 = (const float*)d_in[6];

  const int n = in_sizes[0] / D;
  const int E = in_sizes[2];
  const int* src = ei;        // edge_index[0,:]
  const int* dst = ei + E;    // edge_index[1,:]

  float* ws  = (float*)d_ws;
  float* deg = ws;                         // [n]
  float* dis = ws + n;                     // [n]
  float* xm  = ws + 2 * (size_t)n;         // [n, D]
  float* h   = xm + (size_t)n * D;         // [n, D]
  float* out = (float*)d_out;              // [n, D]

  const int T = 256;  // 8 waves/block on wave32
  const int nBlocks   = (n + T - 1) / T;
  const int eBlocks   = (E + T - 1) / T;
  const long long nd  = (long long)n * D;
  const int ndBlocks  = (int)((nd + T - 1) / T);
  const int edBlocks  = (int)(((long long)E * D + T - 1) / T);
  const int rowsPerBlock = (T / 32) * 16;  // 128 rows per block
  const int gBlocks   = (n + rowsPerBlock - 1) / rowsPerBlock;

  // Normalization coefficients
  k_init_deg<<<nBlocks, T, 0, stream>>>(deg, n);
  k_deg_edges<<<eBlocks, T, 0, stream>>>(dst, ew, deg, E);
  k_dis<<<nBlocks, T, 0, stream>>>(deg, dis, n);

  // Layer 1: h = segment_sum(msg) + b1   (ReLU deferred into layer-2 GEMM read)
  k_gemm32<false><<<gBlocks, T, 0, stream>>>(x, W1, xm, n);
  k_self_bias<<<ndBlocks, T, 0, stream>>>(xm, dis, b1, h, nd);
  k_edge_msg<<<edBlocks, T, 0, stream>>>(src, dst, ew, dis, xm, h, E);

  // Layer 2: out = segment_sum(msg(relu(h))) + b2
  k_gemm32<true><<<gBlocks, T, 0, stream>>>(h, W2, xm, n);
  k_self_bias<<<ndBlocks, T, 0, stream>>>(xm, dis, b2, out, nd);
  k_edge_msg<<<edBlocks, T, 0, stream>>>(src, dst, ew, dis, xm, out, E);
}